// SAGEConv_1554778161245
// MI455X (gfx1250) — compile-verified
//
#include <hip/hip_runtime.h>
#include <hip/hip_bf16.h>

// SAGEConv for MI455X (gfx1250, wave32).
//   agg = scatter_mean(x[row] -> col);  out = x@Wself^T + agg@Wneigh^T + bias
// Phase 1 (dominant, memory/atomic bound): per-edge gather + global_atomic_add_f32.
// Phase 2 (2.6 GFLOP, precision-sensitive): fp32 WMMA V_WMMA_F32_16X16X4_F32.

#define D 128

typedef __attribute__((ext_vector_type(2))) float v2f;
typedef __attribute__((ext_vector_type(8))) float v8f;

// ---------------------------------------------------------------- zero ws ---
__global__ void sage_zero(float* __restrict__ p, int total) {
    int i = blockIdx.x * blockDim.x + threadIdx.x;
    if (i < total) p[i] = 0.0f;
}

// ------------------------------------------------------- edge scatter-add ---
// One wave per edge: each lane moves 4 consecutive floats of the 128-wide row.
// x[src] read as float4 (global_load_b128), accumulated into agg[dst] with
// global_atomic_add_f32 (hot in L2: agg is 20.5 MB << 192 MB L2).
__global__ void sage_scatter(const float* __restrict__ x,
                             const int*   __restrict__ edge_index, // [2, E]
                             float*       __restrict__ agg,
                             float*       __restrict__ deg,
                             int nedges) {
    const int lane = threadIdx.x & 31;
    const int e    = blockIdx.x * (blockDim.x >> 5) + (threadIdx.x >> 5);
    if (e >= nedges) return;

    const int src = edge_index[e];            // row: gather source
    const int dst = edge_index[nedges + e];   // col: scatter destination

    const float4 v = *(const float4*)(x + (size_t)src * D + lane * 4);
    float* dp = agg + (size_t)dst * D + lane * 4;
    atomicAdd(dp + 0, v.x);
    atomicAdd(dp + 1, v.y);
    atomicAdd(dp + 2, v.z);
    atomicAdd(dp + 3, v.w);
    if (lane == 0) atomicAdd(deg + dst, 1.0f);
}

// ------------------------------------------------------------ WMMA GEMM -----
// One wave per 16x16 output tile. D[m][n] = sum_k A[m][k]*B[k][n] with
//   A = x (and agg*invdeg),  B[k][n] = W[n*128 + k]  (W row-major [out,in]).
// Fragment layout (ISA 7.12.2, f32 16x16x4):
//   A: lane l -> row m=l&15, VGPR v -> K = (l>>4)*2 + v   (float2 per lane)
//   B: lane l -> col n=l&15, VGPR v -> K = (l>>4)*2 + v   (float2 per lane)
//   C/D: VGPR v -> M = v + ((l>>4)<<3), N = l&15          (8 floats per lane)
__global__ void sage_gemm(const float* __restrict__ x,
                          const float* __restrict__ agg,
                          const float* __restrict__ deg,
                          const float* __restrict__ Wself,
                          const float* __restrict__ Wneigh,
                          const float* __restrict__ bias,
                          float* __restrict__ out,
                          int mtiles) {             // nodes/16
    const int lane  = threadIdx.x & 31;
    const int tile  = blockIdx.x * (blockDim.x >> 5) + (threadIdx.x >> 5);
    if (tile >= mtiles * (D / 16)) return;          // whole wave uniform branch
    const int mtile = tile >> 3;                    // D/16 == 8 N-tiles
    const int ntile = tile & 7;

    const int l15   = lane & 15;
    const int kbase = (lane >> 4) << 1;             // 0 or 2
    const int m     = (mtile << 4) + l15;           // A-fragment row
    const int n     = (ntile << 4) + l15;           // B-fragment col

    const float invd = 1.0f / fmaxf(deg[m], 1.0f);  // fold mean-normalization
    const float b    = bias[n];

    v8f acc;                                        // init C with bias[n]
#pragma unroll
    for (int v = 0; v < 8; ++v) acc[v] = b;

    const float* xrow = x      + (size_t)m * D + kbase;
    const float* arow = agg    + (size_t)m * D + kbase;
    const float* wsr  = Wself  + (size_t)n * D + kbase;  // B_self[k][n]
    const float* wnr  = Wneigh + (size_t)n * D + kbase;  // B_neigh[k][n]

#pragma unroll 4
    for (int k = 0; k < D; k += 4) {
        v2f a  = *(const v2f*)(xrow + k);
        v2f bs = *(const v2f*)(wsr + k);
        acc = __builtin_amdgcn_wmma_f32_16x16x4_f32(
            false, a, false, bs, (short)0, acc, false, false);

        v2f a2 = *(const v2f*)(arow + k);
        a2 = a2 * invd;                              // scatter-mean scaling
        v2f bn = *(const v2f*)(wnr + k);
        acc = __builtin_amdgcn_wmma_f32_16x16x4_f32(
            false, a2, false, bn, (short)0, acc, false, false);
    }

    // Write back D-matrix per C/D layout.
    const int mrow0 = (mtile << 4) + ((lane >> 4) << 3);
    const int ncol  = (ntile << 4) + l15;
#pragma unroll
    for (int v = 0; v < 8; ++v)
        out[(size_t)(mrow0 + v) * D + ncol] = acc[v];
}

// ----------------------------------------------------------------- driver ---
extern "C" void kernel_launch(void* const* d_in, const int* in_sizes, int n_in,
                              void* d_out, int out_size, void* d_ws, size_t ws_size,
                              hipStream_t stream) {
    const float* x      = (const float*)d_in[0];
    const int*   eidx   = (const int*)  d_in[1];   // [2, E]
    const float* Wself  = (const float*)d_in[2];
    const float* Wneigh = (const float*)d_in[3];
    const float* bias   = (const float*)d_in[4];
    float*       out    = (float*)d_out;

    const int nnodes = in_sizes[0] / D;            // 40000
    const int nedges = in_sizes[1] / 2;            // 640000

    float* agg = (float*)d_ws;                     // [nnodes, D]
    float* deg = agg + (size_t)nnodes * D;         // [nnodes]

    // 1) zero agg + deg
    const int ztotal = nnodes * D + nnodes;
    sage_zero<<<(ztotal + 255) / 256, 256, 0, stream>>>(agg, ztotal);

    // 2) scatter-add: one wave per edge, 8 edges per 256-thread block
    sage_scatter<<<(nedges + 7) / 8, 256, 0, stream>>>(x, eidx, agg, deg, nedges);

    // 3) fused normalize + dual GEMM + bias via fp32 WMMA
    const int mtiles = nnodes / 16;                // 2500 (exact)
    const int tiles  = mtiles * (D / 16);          // 20000 waves
    sage_gemm<<<(tiles + 7) / 8, 256, 0, stream>>>(x, agg, deg, Wself, Wneigh,
                                                   bias, out, mtiles);
}